// GaussPSF_6262062317778
// MI455X (gfx1250) — compile-verified
//
#include <hip/hip_runtime.h>
#include <stdint.h>

// Problem constants (from reference)
#define Bn 8
#define Cn 3
#define Hn 768
#define Wn 768
#define Rr 3            // K=7 -> radius 3

// Tiling
#define TH 16
#define TW 64
#define HALO_H (TH + 2*Rr)        // 22 rows
#define HALO_W (TW + 2*Rr + 2)    // 72 cols (base col = w0-4, one spare col each side)
#define LDS_STRIDE 73             // odd stride: lanes 0-15 / 16-31 hit disjoint LDS banks

__device__ __forceinline__ int iclamp(int v, int lo, int hi) {
    return v < lo ? lo : (v > hi ? hi : v);
}

__global__ __launch_bounds__(256) void gpsf_kernel(
    const float* __restrict__ img,   // [B,C,H,W]
    const float* __restrict__ dep,   // [B,H,W]
    const float* __restrict__ fdp,   // [B]
    const float* __restrict__ app,   // [B]
    const float* __restrict__ flp,   // [B]
    float* __restrict__ out)         // [B,C,H,W]
{
    __shared__ float s_tile[HALO_H * LDS_STRIDE];          // per-pixel exp2 slope s = -log2e/(2c^2)
    __shared__ float img_tile[Cn][HALO_H * LDS_STRIDE];    // 3 channel halo tiles

    const int b  = blockIdx.z;
    const int h0 = blockIdx.y * TH;
    const int w0 = blockIdx.x * TW;
    const int tid = threadIdx.x;

    // Per-batch optics (uniform -> scalar loads)
    const float fd = fdp[b], ap = app[b], FL = flp[b];
    const float rfd    = 9.0f * fd + 1.0f;                       // (FAR-NEAR)*fd + NEAR
    const float rfd_FL = rfd - FL;
    const float K1     = FL * __builtin_amdgcn_rcpf(ap) * FL *
                         (1.0f / (5.6e-06f * 3.0f));             // Ap*FL / (PIXEL_SIZE*SCALE)

    // ---------------- Tile stage ----------------
    // depth -> s via VALU + ds_store; image channels via gfx1250 async global->LDS copies.
    for (int i = tid; i < HALO_H * HALO_W; i += 256) {
        const int r  = i / HALO_W;
        const int cc = i - r * HALO_W;
        const int hh = h0 - Rr + r;
        const int ww = w0 - (Rr + 1) + cc;
        const int hc = iclamp(hh, 0, Hn - 1);
        const int wc = iclamp(ww, 0, Wn - 1);
        const bool oob = (hh != hc) | (ww != wc);
        const int l = r * LDS_STRIDE + cc;

        // circle of confusion -> exp2 slope
        const float d   = dep[(size_t)(b * Hn + hc) * Wn + wc];
        const float rd  = 9.0f * d + 1.0f;
        const float coc = fabsf(K1 * (rd - rfd) * __builtin_amdgcn_rcpf(rd * rfd_FL));
        const float c   = fminf(fmaxf(coc, 1.0f), 7.0f);
        float s = -0.72134752044f * __builtin_amdgcn_rcpf(c * c); // -log2(e)/2 / c^2
        if (oob) s = -1.0e30f;   // weight == 0 for every r2 >= 1 tap (matches zero padding)
        s_tile[l] = s;

        // Async copy the 3 image channels for this halo pixel (per-lane gathered copy).
        const uint32_t la0 = (uint32_t)(uintptr_t)(&img_tile[0][l]);
        #pragma unroll
        for (int ch = 0; ch < Cn; ++ch) {
            const uint64_t ga =
                (uint64_t)(uintptr_t)(img + ((size_t)(b * Cn + ch) * Hn + hc) * Wn + wc);
            const uint32_t la = la0 + (uint32_t)(ch * HALO_H * LDS_STRIDE * 4);
            asm volatile("global_load_async_to_lds_b32 %0, %1, off"
                         :: "v"(la), "v"(ga) : "memory");
        }
    }
    asm volatile("s_wait_asynccnt 0x0" ::: "memory");  // this wave's async copies done
    __syncthreads();                                   // all waves' copies + s stores visible

    // ---------------- Compute stage ----------------
    // Thread -> 1x4 output strip: row ty, cols cbase..cbase+3 (tile coords).
    const int tx = tid & 15;
    const int ty = tid >> 4;
    const int cbase = tx * 4;

    float den[4]    = {0.0f, 0.0f, 0.0f, 0.0f};
    float num[Cn][4] = {};

    #pragma unroll
    for (int dy = -Rr; dy <= Rr; ++dy) {
        const int   lr  = (ty + Rr + dy) * LDS_STRIDE;
        const float dy2 = (float)(dy * dy);

        // Row working set: LDS cols cbase+1 .. cbase+10  (center of pixel p is cbase+4+p)
        float sr[10], ir[Cn][10];
        #pragma unroll
        for (int j = 0; j < 10; ++j) {
            const int l = lr + cbase + 1 + j;
            sr[j]    = s_tile[l];
            ir[0][j] = img_tile[0][l];
            ir[1][j] = img_tile[1][l];
            ir[2][j] = img_tile[2][l];
        }
        #pragma unroll
        for (int dx = -Rr; dx <= Rr; ++dx) {
            const float r2 = dy2 + (float)(dx * dx);
            #pragma unroll
            for (int p = 0; p < 4; ++p) {
                const int j = 3 + p + dx;                       // 0..9
                const float w = __builtin_amdgcn_exp2f(sr[j] * r2); // v_exp_f32
                den[p]   += w;
                num[0][p] = fmaf(w, ir[0][j], num[0][p]);
                num[1][p] = fmaf(w, ir[1][j], num[1][p]);
                num[2][p] = fmaf(w, ir[2][j], num[2][p]);
            }
        }
    }

    // Normalize: rcp + one Newton step (den >= 1 always: center tap weight is 1).
    float inv[4];
    #pragma unroll
    for (int p = 0; p < 4; ++p) {
        float iv = __builtin_amdgcn_rcpf(den[p]);
        inv[p] = iv * (2.0f - den[p] * iv);
    }

    const int oh = h0 + ty;
    const int ow = w0 + cbase;
    #pragma unroll
    for (int ch = 0; ch < Cn; ++ch) {
        float4 v;
        v.x = num[ch][0] * inv[0];
        v.y = num[ch][1] * inv[1];
        v.z = num[ch][2] * inv[2];
        v.w = num[ch][3] * inv[3];
        *(float4*)(out + ((size_t)(b * Cn + ch) * Hn + oh) * Wn + ow) = v;
    }
}

extern "C" void kernel_launch(void* const* d_in, const int* in_sizes, int n_in,
                              void* d_out, int out_size, void* d_ws, size_t ws_size,
                              hipStream_t stream) {
    const float* image = (const float*)d_in[0];
    const float* depth = (const float*)d_in[1];
    const float* fdp   = (const float*)d_in[2];
    const float* app   = (const float*)d_in[3];
    const float* flp   = (const float*)d_in[4];
    float* out = (float*)d_out;

    dim3 grid(Wn / TW, Hn / TH, Bn);   // 12 x 48 x 8
    gpsf_kernel<<<grid, 256, 0, stream>>>(image, depth, fdp, app, flp, out);
}